// NonSelectiveSSMKernel_33002528702786
// MI455X (gfx1250) — compile-verified
//
#include <hip/hip_runtime.h>

typedef float v2f __attribute__((ext_vector_type(2)));
typedef float v8f __attribute__((ext_vector_type(8)));

#define N_STATE 64
#define CHQ     64        // chunk length
#define NCHUNK  32        // 2048 / 64
#define NCH     2048      // B * d_model channels
#define LSEQ    2048
#define LDP     65        // padded LDS row stride

#define AS1 __attribute__((address_space(1)))
#define AS3 __attribute__((address_space(3)))

// ---------------------------------------------------------------------------
// gfx1250 async global->LDS copy (4 bytes per lane, per-lane LDS address),
// tracked by ASYNCcnt.  Builtin signature (probe-confirmed by diagnostics):
//   (AS1 int* src, AS3 int* dst, imm offset, imm cpol)
// LDS byte offset == low 32 bits of the generic pointer (ISA aperture rule).
// ---------------------------------------------------------------------------
__device__ __forceinline__ void async_g2l(const float* g, float* l) {
#if defined(__has_builtin) && __has_builtin(__builtin_amdgcn_global_load_async_to_lds_b32)
  __builtin_amdgcn_global_load_async_to_lds_b32(
      (AS1 int*)(unsigned long long)g,
      (AS3 int*)(unsigned)(unsigned long long)l, 0, 0);
#else
  asm volatile("global_load_async_to_lds_b32 %0, %1, off"
               :: "v"((unsigned)(unsigned long long)l), "v"(g)
               : "memory");
#endif
}

#if defined(__has_builtin) && __has_builtin(__builtin_amdgcn_s_wait_asynccnt)
#define WAIT_ASYNC(n) __builtin_amdgcn_s_wait_asynccnt(n)
#else
#define WAIT_ASYNC(n) asm volatile("s_wait_asynccnt %0" :: "i"(n) : "memory")
#endif

// ---------------------------------------------------------------------------
// 64x64x64 fp32 GEMM accumulate using v_wmma_f32_16x16x4_f32.
// A,B tiles live in LDS as [64][LDP] row-major (A: MxK, B: KxN).
// Each wave (of 4) computes rows [wave*16, wave*16+16) x all 64 cols.
// ---------------------------------------------------------------------------
__device__ __forceinline__ void gemm64_accum(const float* Alds, const float* Blds,
                                             v8f acc[4], int wave, int lane) {
  const int lr = lane & 15;            // row (A) / col (B) within tile
  const int lh = (lane >> 4) << 1;     // K sub-offset: 0 (lanes 0-15) or 2 (16-31)
  const int rowbase = wave * 16;
#pragma unroll
  for (int kk = 0; kk < 64; kk += 4) {
    v2f a;
    a.x = Alds[(rowbase + lr) * LDP + kk + lh + 0];
    a.y = Alds[(rowbase + lr) * LDP + kk + lh + 1];
#pragma unroll
    for (int j = 0; j < 4; ++j) {
      v2f b;
      b.x = Blds[(kk + lh + 0) * LDP + j * 16 + lr];
      b.y = Blds[(kk + lh + 1) * LDP + j * 16 + lr];
      acc[j] = __builtin_amdgcn_wmma_f32_16x16x4_f32(
          /*neg_a=*/false, a, /*neg_b=*/false, b,
          /*c_mod=*/(short)0, acc[j], /*reuse_a=*/false, /*reuse_b=*/false);
    }
  }
}

// ---------------------------------------------------------------------------
// Kernel 0: build M (Q x Q Toeplitz), W (N x Q), V (Q x N), apow (a^Q)
// from diag(A), Bvec, Cvec.  One block of 64 threads.
// ---------------------------------------------------------------------------
__global__ void ssm_precompute(const float* __restrict__ A,
                               const float* __restrict__ Bv,
                               const float* __restrict__ Cv,
                               float* __restrict__ Mg, float* __restrict__ Wg,
                               float* __restrict__ Vg, float* __restrict__ apowg) {
  __shared__ float pw[64][66];   // a_i^k, k = 0..64
  __shared__ float cb[64];
  __shared__ float Ks[64];
  const int i = threadIdx.x;     // state index, 64 threads
  const float a = A[i * N_STATE + i];   // A is diagonal by construction
  float p = 1.f;
  for (int k = 0; k <= 64; ++k) { pw[i][k] = p; p *= a; }
  cb[i] = Cv[i] * Bv[i];
  __syncthreads();

  apowg[i] = pw[i][64];
  const float bi = Bv[i], ci = Cv[i];
  for (int s = 0; s < CHQ; ++s) Wg[i * CHQ + s] = bi * pw[i][63 - s];
  for (int t = 0; t < CHQ; ++t) Vg[t * N_STATE + i] = ci * pw[i][t + 1];

  float K = 0.f;                       // thread i computes K[i] = sum_j C_j B_j a_j^i
  for (int j = 0; j < 64; ++j) K += cb[j] * pw[j][i];
  Ks[i] = K;
  __syncthreads();
  for (int t = 0; t < CHQ; ++t)        // thread i writes column s = i
    Mg[t * CHQ + i] = (t >= i) ? Ks[t - i] : 0.f;
}

// ---------------------------------------------------------------------------
// Kernel A: Z[c] = W @ U[c]   (per chunk c, per 64-channel tile)
// grid = (32 chunks, 32 channel tiles), block = 128 (4 waves)
// ---------------------------------------------------------------------------
__global__ void ssm_gemm_z(const float* __restrict__ u, const float* __restrict__ Wg,
                           float* __restrict__ ZH) {
  __shared__ float sA[64 * LDP];
  __shared__ float sB[64 * LDP];
  const int tid = threadIdx.x;
  const int chunk = blockIdx.x;
  const int chbase = blockIdx.y * 64;
  const int cbase = chunk * CHQ;

  // Async-stage W[i][s] and U[s][ch] (transposed) straight into LDS.
  for (int idx = tid; idx < 4096; idx += 128) {
    const int rr = idx >> 6, cc = idx & 63;
    async_g2l(Wg + idx, &sA[rr * LDP + cc]);
    // consecutive tid -> consecutive cc == consecutive time s (unit stride in u)
    async_g2l(u + (size_t)(chbase + rr) * LSEQ + cbase + cc, &sB[cc * LDP + rr]);
  }
  WAIT_ASYNC(0);
  __syncthreads();

  const int lane = tid & 31, wave = tid >> 5;
  v8f acc[4] = {};
  gemm64_accum(sA, sB, acc, wave, lane);

  const int lr = lane & 15, rofs = (lane >> 4) * 8, rowbase = wave * 16;
#pragma unroll
  for (int j = 0; j < 4; ++j)
#pragma unroll
    for (int r = 0; r < 8; ++r)
      ZH[(size_t)(chunk * N_STATE + rowbase + rofs + r) * NCH + chbase + j * 16 + lr] =
          acc[j][r];
}

// ---------------------------------------------------------------------------
// Kernel B: in-place chunk-level scan.  On entry ZH[c] = Z_c; on exit
// ZH[c] = H_c (state entering chunk c):  H_0 = 0, H_c = a^Q*H_{c-1} + Z_{c-1}.
// One thread per (state, channel) = 131072 threads.
// ---------------------------------------------------------------------------
__global__ void ssm_scan(float* __restrict__ ZH, const float* __restrict__ apow) {
  const int g = blockIdx.x * blockDim.x + threadIdx.x;
  const int i = g >> 11;        // state 0..63
  const int ch = g & (NCH - 1);
  const float ap = apow[i];
  float h = 0.f;
  for (int c = 0; c < NCHUNK; ++c) {
    const size_t idx = ((size_t)(c * N_STATE + i) << 11) + ch;
    const float z = ZH[idx];
    ZH[idx] = h;
    h = ap * h + z;
  }
}

// ---------------------------------------------------------------------------
// Kernel C: Y[c] = V @ H_c + M @ U[c]
// grid = (32 chunks, 32 channel tiles), block = 128 (4 waves)
// V,H staged first; U issued behind them and left in flight under GEMM-1
// (in-order async completion: cnt<=32 after 96 issues => all 64 V/H done).
// ---------------------------------------------------------------------------
__global__ void ssm_gemm_y(const float* __restrict__ u, const float* __restrict__ Mg,
                           const float* __restrict__ Vg, const float* __restrict__ ZH,
                           float* __restrict__ y) {
  __shared__ float sA[64 * LDP];   // V, later M
  __shared__ float sB[64 * LDP];   // H, later Y staging
  __shared__ float sC[64 * LDP];   // U
  const int tid = threadIdx.x;
  const int chunk = blockIdx.x;
  const int chbase = blockIdx.y * 64;
  const int cbase = chunk * CHQ;

  for (int idx = tid; idx < 4096; idx += 128) {
    const int rr = idx >> 6, cc = idx & 63;
    async_g2l(Vg + idx, &sA[rr * LDP + cc]);                                  // V[t][i]
    async_g2l(ZH + (size_t)(chunk * N_STATE + rr) * NCH + chbase + cc,
              &sB[rr * LDP + cc]);                                            // H[i][ch]
  }
  for (int idx = tid; idx < 4096; idx += 128) {
    const int rr = idx >> 6, cc = idx & 63;
    async_g2l(u + (size_t)(chbase + rr) * LSEQ + cbase + cc,
              &sC[cc * LDP + rr]);                                            // U[s][ch]
  }
  WAIT_ASYNC(32);   // first 64 per-thread issues (V,H) complete; U may be in flight
  __syncthreads();

  const int lane = tid & 31, wave = tid >> 5;
  v8f acc[4] = {};
  gemm64_accum(sA, sB, acc, wave, lane);      // acc = V @ H
  __syncthreads();                            // all waves done reading sA

  for (int idx = tid; idx < 4096; idx += 128) {
    const int rr = idx >> 6, cc = idx & 63;
    async_g2l(Mg + idx, &sA[rr * LDP + cc]);                                  // M[t][s]
  }
  WAIT_ASYNC(0);    // U and M fully resident
  __syncthreads();

  gemm64_accum(sA, sC, acc, wave, lane);      // acc += M @ U
  __syncthreads();

  // Stage Y tile in LDS so the global stores coalesce along the time axis.
  const int lr = lane & 15, rofs = (lane >> 4) * 8, rowbase = wave * 16;
#pragma unroll
  for (int j = 0; j < 4; ++j)
#pragma unroll
    for (int r = 0; r < 8; ++r)
      sB[(rowbase + rofs + r) * LDP + j * 16 + lr] = acc[j][r];
  __syncthreads();

  for (int idx = tid; idx < 4096; idx += 128) {
    const int ch = idx >> 6, t = idx & 63;
    y[(size_t)(chbase + ch) * LSEQ + cbase + t] = sB[t * LDP + ch];
  }
}

// ---------------------------------------------------------------------------
extern "C" void kernel_launch(void* const* d_in, const int* in_sizes, int n_in,
                              void* d_out, int out_size, void* d_ws, size_t ws_size,
                              hipStream_t stream) {
  const float* u  = (const float*)d_in[0];   // (4, 512, 2048)
  const float* A  = (const float*)d_in[1];   // (64, 64) diagonal
  const float* Bv = (const float*)d_in[2];   // (64,)
  const float* Cv = (const float*)d_in[3];   // (64,)
  // d_in[4] = L (scalar), fixed at 2048

  float* ws   = (float*)d_ws;
  float* Mg   = ws;                 // 4096
  float* Wg   = ws + 4096;          // 4096
  float* Vg   = ws + 8192;          // 4096
  float* apow = ws + 12288;         // 64
  float* ZH   = ws + 12352;         // 32*64*2048 = 4,194,304 floats

  float* y = (float*)d_out;

  ssm_precompute<<<1, 64, 0, stream>>>(A, Bv, Cv, Mg, Wg, Vg, apow);
  ssm_gemm_z<<<dim3(NCHUNK, NCH / 64), 128, 0, stream>>>(u, Wg, ZH);
  ssm_scan<<<(N_STATE * NCH) / 256, 256, 0, stream>>>(ZH, apow);
  ssm_gemm_y<<<dim3(NCHUNK, NCH / 64), 128, 0, stream>>>(u, Mg, Vg, ZH, y);
}